// CCC_Loss_45646912422065
// MI455X (gfx1250) — compile-verified
//
#include <hip/hip_runtime.h>

// ---------------------------------------------------------------------------
// CCC loss, MI455X (gfx1250, wave32).
// Memory-bound streaming reduction: 256 MB @ 23.3 TB/s ~= 11 us floor.
// Phase 1: 4096 blocks x 256 threads, coalesced NT float4 loads, 5 running
//          sums per thread; intra-wave fold uses V_WMMA_F32_16X16X4_F32
//          (A = ones => D[m][n] = sum_k B[k][n], i.e. lane n + lane n+16),
//          then 4 shfl_xor steps finish the 32-lane sum.
// Phase 2: 1 block finalizes per-row CCC and the mean over rows.
// ---------------------------------------------------------------------------

typedef __attribute__((ext_vector_type(2))) float v2f;
typedef __attribute__((ext_vector_type(4))) float v4f;
typedef __attribute__((ext_vector_type(8))) float v8f;

#define T_LEN   65536
#define B_ROWS  512
#define SPLIT   8                      // segments per row
#define CHUNK   (T_LEN / SPLIT)        // 8192 elements per block per array
#define THREADS 256
#define WAVES   (THREADS / 32)
#define VEC_IT  (CHUNK / (THREADS * 4))// 8 float4 iterations per thread

// Full 32-lane sum, returned to every lane. One v_wmma_f32_16x16x4_f32 folds
// lane n with lane n+16 across all 16 columns (ones-matrix sums all K slots,
// so this is layout-robust); 4 shfl_xor steps then sum the 16 columns, which
// equals the total of all 32 input lanes.
__device__ __forceinline__ float wave_sum32(float s) {
    v2f a; a[0] = 1.0f; a[1] = 1.0f;     // A = ones(16x4)
    v2f b; b[0] = s;    b[1] = 0.0f;     // B carries one value per lane
    v8f c = {};
    v8f d = __builtin_amdgcn_wmma_f32_16x16x4_f32(
        /*neg_a=*/false, a, /*neg_b=*/false, b,
        /*c_mod=*/(short)0, c, /*reuse_a=*/false, /*reuse_b=*/false);
    float t = d[0];                      // row 0 of D: per-column K-sums
    t += __shfl_xor(t, 1, 32);
    t += __shfl_xor(t, 2, 32);
    t += __shfl_xor(t, 4, 32);
    t += __shfl_xor(t, 8, 32);
    return t;
}

__global__ __launch_bounds__(THREADS)
void ccc_partial_kernel(const float* __restrict__ preds,
                        const float* __restrict__ labels,
                        float* __restrict__ part) {
    const int blk = blockIdx.x;
    const int row = blk / SPLIT;
    const int seg = blk % SPLIT;
    const int tid = threadIdx.x;

    const size_t base = (size_t)row * T_LEN + (size_t)seg * CHUNK;
    const v4f* xp = (const v4f*)(preds + base);
    const v4f* yp = (const v4f*)(labels + base);

    float sx = 0.f, sy = 0.f, sxx = 0.f, syy = 0.f, sxy = 0.f;

#pragma unroll
    for (int it = 0; it < VEC_IT; ++it) {
        // Coalesced: 256 threads x 16 B = 4 KB contiguous per iteration.
        // NT hint: 256 MB stream > 192 MB L2, don't pollute.
        v4f x = __builtin_nontemporal_load(&xp[it * THREADS + tid]);
        v4f y = __builtin_nontemporal_load(&yp[it * THREADS + tid]);
#pragma unroll
        for (int k = 0; k < 4; ++k) {
            float xv = x[k], yv = y[k];
            sx  += xv;
            sy  += yv;
            sxx  = fmaf(xv, xv, sxx);
            syy  = fmaf(yv, yv, syy);
            sxy  = fmaf(xv, yv, sxy);
        }
    }

    // Intra-wave 32-lane reduction via WMMA + shuffles (EXEC is all-1s here:
    // uniform control flow, every wave executes).
    sx  = wave_sum32(sx);
    sy  = wave_sum32(sy);
    sxx = wave_sum32(sxx);
    syy = wave_sum32(syy);
    sxy = wave_sum32(sxy);

    __shared__ float wsum[WAVES][5];
    const int wave = tid >> 5;
    if ((tid & 31) == 0) {
        wsum[wave][0] = sx;  wsum[wave][1] = sy;
        wsum[wave][2] = sxx; wsum[wave][3] = syy;
        wsum[wave][4] = sxy;
    }
    __syncthreads();

    if (tid == 0) {
        float o[5] = {0.f, 0.f, 0.f, 0.f, 0.f};
#pragma unroll
        for (int w = 0; w < WAVES; ++w)
#pragma unroll
            for (int q = 0; q < 5; ++q)
                o[q] += wsum[w][q];
        float* dst = part + (size_t)blk * 5;
#pragma unroll
        for (int q = 0; q < 5; ++q) dst[q] = o[q];
    }
}

__global__ __launch_bounds__(B_ROWS)
void ccc_finalize_kernel(const float* __restrict__ part,
                         float* __restrict__ out) {
    const int r = threadIdx.x;           // one thread per row, 512 threads

    float sx = 0.f, sy = 0.f, sxx = 0.f, syy = 0.f, sxy = 0.f;
    const float* p = part + (size_t)r * SPLIT * 5;
#pragma unroll
    for (int s = 0; s < SPLIT; ++s) {
        sx  += p[s * 5 + 0];
        sy  += p[s * 5 + 1];
        sxx += p[s * 5 + 2];
        syy += p[s * 5 + 3];
        sxy += p[s * 5 + 4];
    }

    const float invT = 1.0f / (float)T_LEN;
    float mx  = sx * invT;
    float my  = sy * invT;
    float vx  = sxx * invT - mx * mx;
    float vy  = syy * invT - my * my;
    float cxy = sxy * invT - mx * my;
    float dm  = mx - my;
    float ccc = (2.0f * cxy) / (vx + vy + dm * dm + 1e-8f);

    __shared__ float red[B_ROWS];
    red[r] = 1.0f - ccc;
    __syncthreads();
#pragma unroll
    for (int off = B_ROWS / 2; off > 0; off >>= 1) {
        if (r < off) red[r] += red[r + off];
        __syncthreads();
    }
    if (r == 0) out[0] = red[0] * (1.0f / (float)B_ROWS);
}

extern "C" void kernel_launch(void* const* d_in, const int* in_sizes, int n_in,
                              void* d_out, int out_size, void* d_ws, size_t ws_size,
                              hipStream_t stream) {
    (void)in_sizes; (void)n_in; (void)out_size; (void)ws_size;
    const float* preds  = (const float*)d_in[0];
    const float* labels = (const float*)d_in[1];
    float* part = (float*)d_ws;          // 512 * 8 * 5 floats = 80 KB scratch

    ccc_partial_kernel<<<B_ROWS * SPLIT, THREADS, 0, stream>>>(preds, labels, part);
    ccc_finalize_kernel<<<1, B_ROWS, 0, stream>>>(part, (float*)d_out);
}